// Smpl_51256139710911
// MI455X (gfx1250) — compile-verified
//
#include <hip/hip_runtime.h>

// ---------------- problem dims ----------------
#define B_   1024
#define V_   6890
#define J_   24
#define NB_  10
#define NP_  207
#define NK_  19
#define N3   (V_*3)      // 20670
#define INS  82          // inputs row stride (72 pose + 10 shape)
#define KTOT 217         // 10 (shape) + 207 (pose feature)
#define KPAD 220         // padded to multiple of 4 for wmma_f32_16x16x4
#define LDA  224         // LDS row stride (floats)

// ---------------- workspace layout (float offsets) ----------------
#define OFF_ASTAGE  ((size_t)0)                       // B*KPAD          = 225280
#define OFF_JD      ((size_t)(B_*KPAD))               // NB*J*3          = 720
#define OFF_JT      (OFF_JD + NB_*J_*3)               // J*3             = 72
#define OFF_RELA    ((size_t)226080)                  // B*J*16          = 393216
#define OFF_DIRS    ((size_t)619520)                  // (KPAD/2)*2*N3   = 4547400
#define OFF_VPOSED  ((size_t)5167360)                 // B*N3            = 21166080

// ---------------- output layout (float offsets) ----------------
#define OUT_VERT ((size_t)0)                          // B*V*3
#define OUT_JNT  ((size_t)(B_*(size_t)V_*3))          // B*NK*3
#define OUT_RS   (OUT_JNT + (size_t)B_*NK_*3)         // B*J*9

typedef __attribute__((ext_vector_type(2))) float v2f;
typedef __attribute__((ext_vector_type(8))) float v8f;

// =====================================================================
// 1) Build padded A matrix [B, KPAD]: cols 0..9 = shape betas, 217..219 = 0
// =====================================================================
__global__ void k_init_astage(const float* __restrict__ inputs,
                              float* __restrict__ ws) {
    int t = blockIdx.x * blockDim.x + threadIdx.x;
    if (t >= B_ * KPAD) return;
    int b = t / KPAD, k = t % KPAD;
    float v = (k < NB_) ? inputs[(size_t)b * INS + 72 + k] : 0.0f;
    ws[OFF_ASTAGE + (size_t)b * KPAD + k] = v;
}

// =====================================================================
// 2) Rodrigues per (b,j): R -> out Rs; (R - I) -> A cols 10..216
// =====================================================================
__global__ void k_rodrigues(const float* __restrict__ inputs,
                            float* __restrict__ ws,
                            float* __restrict__ out) {
    int t = blockIdx.x * blockDim.x + threadIdx.x;
    if (t >= B_ * J_) return;
    int b = t / J_, j = t % J_;
    const float* aa = inputs + (size_t)b * INS + j * 3;
    float x = aa[0], y = aa[1], z = aa[2];
    float ang = sqrtf(x * x + y * y + z * z + 1e-8f);
    float inv = 1.0f / ang;
    x *= inv; y *= inv; z *= inv;
    float c = cosf(ang), s = sinf(ang), mc = 1.0f - c;
    float R[9];
    R[0] = c + mc * x * x;  R[1] = mc * x * y - s * z; R[2] = mc * x * z + s * y;
    R[3] = mc * x * y + s * z; R[4] = c + mc * y * y;  R[5] = mc * y * z - s * x;
    R[6] = mc * x * z - s * y; R[7] = mc * y * z + s * x; R[8] = c + mc * z * z;

    float* rs = out + OUT_RS + (size_t)b * (J_ * 9) + j * 9;
#pragma unroll
    for (int i = 0; i < 9; ++i) rs[i] = R[i];

    if (j > 0) {
        float* pf = ws + OFF_ASTAGE + (size_t)b * KPAD + 10 + (j - 1) * 9;
#pragma unroll
        for (int i = 0; i < 9; ++i) pf[i] = R[i] - ((i % 4 == 0) ? 1.0f : 0.0f);
    }
}

// =====================================================================
// 3) Pack B matrix (Dirs) into K-pair-interleaved layout so the WMMA
//    B fragment is a single b64 load:
//    Dp[(k>>1)*(2*N3) + 2*n + (k&1)] = Dirs[k][n]
//    rows 0..9 = shapedirs, 10..216 = posedirs, 217..219 = 0
// =====================================================================
__global__ void k_pack_dirs(const float* __restrict__ shp,
                            const float* __restrict__ pos,
                            float* __restrict__ ws) {
    const int pairs = KPAD / 2;                  // 110
    int t = blockIdx.x * blockDim.x + threadIdx.x;
    if (t >= pairs * N3) return;
    int p = t / N3, n = t % N3;
    int k0 = 2 * p, k1 = k0 + 1;
    float x = (k0 < NB_)  ? shp[(size_t)k0 * N3 + n]
            : (k0 < KTOT) ? pos[(size_t)(k0 - NB_) * N3 + n] : 0.0f;
    float y = (k1 < NB_)  ? shp[(size_t)k1 * N3 + n]
            : (k1 < KTOT) ? pos[(size_t)(k1 - NB_) * N3 + n] : 0.0f;
    float* d = ws + OFF_DIRS + (size_t)p * (2 * N3) + 2 * (size_t)n;
    d[0] = x; d[1] = y;
}

// =====================================================================
// 4) Joint-regressor precontraction (wave per (n,k), shfl_xor reduce)
// =====================================================================
__global__ void k_joint_reduce(const float* __restrict__ vtpl,
                               const float* __restrict__ shapedirs,
                               const float* __restrict__ jreg,
                               float* __restrict__ ws) {
    int unit = blockIdx.x * (blockDim.x >> 5) + (threadIdx.x >> 5);
    if (unit >= (NB_ + 1) * J_) return;
    int n = unit / J_, k = unit % J_;
    int lane = threadIdx.x & 31;
    const float* src = (n < NB_) ? (shapedirs + (size_t)n * N3) : vtpl;
    float a0 = 0.f, a1 = 0.f, a2 = 0.f;
    for (int v = lane; v < V_; v += 32) {
        float w = jreg[(size_t)v * J_ + k];
        const float* s = src + (size_t)v * 3;
        a0 += w * s[0]; a1 += w * s[1]; a2 += w * s[2];
    }
#pragma unroll
    for (int m = 16; m >= 1; m >>= 1) {
        a0 += __shfl_xor(a0, m, 32);
        a1 += __shfl_xor(a1, m, 32);
        a2 += __shfl_xor(a2, m, 32);
    }
    if (lane == 0) {
        float* dst = (n < NB_) ? (ws + OFF_JD + (size_t)(n * J_ + k) * 3)
                               : (ws + OFF_JT + (size_t)k * 3);
        dst[0] = a0; dst[1] = a1; dst[2] = a2;
    }
}

// =====================================================================
// 5) Per-batch kinematic chain -> rel_A [B,24,16]
// =====================================================================
__global__ void k_chain(const float* __restrict__ inputs,
                        const float* __restrict__ out,   // read Rs region
                        float* __restrict__ ws) {
    int b = blockIdx.x * blockDim.x + threadIdx.x;
    if (b >= B_) return;
    const int PAR[24] = {0,0,0,0,1,2,3,4,5,6,7,8,9,9,9,12,13,14,16,17,18,19,20,21};

    float sh[NB_];
#pragma unroll
    for (int n = 0; n < NB_; ++n) sh[n] = inputs[(size_t)b * INS + 72 + n];

    float Js[J_][3];
#pragma unroll
    for (int k = 0; k < J_; ++k) {
#pragma unroll
        for (int c = 0; c < 3; ++c) {
            float acc = ws[OFF_JT + k * 3 + c];
#pragma unroll
            for (int n = 0; n < NB_; ++n)
                acc += sh[n] * ws[OFF_JD + (size_t)(n * J_ + k) * 3 + c];
            Js[k][c] = acc;
        }
    }

    const float* rsB = out + OUT_RS + (size_t)b * (J_ * 9);
    float res[J_][12];   // global transforms, 3x4 each (bottom row implicit 0001)
    {
        const float* R = rsB;
#pragma unroll
        for (int r = 0; r < 3; ++r) {
#pragma unroll
            for (int c2 = 0; c2 < 3; ++c2) res[0][r * 4 + c2] = R[r * 3 + c2];
            res[0][r * 4 + 3] = Js[0][r];
        }
    }
#pragma unroll
    for (int i = 1; i < J_; ++i) {
        const int p = PAR[i];
        const float* R = rsB + i * 9;
        float rj0 = Js[i][0] - Js[p][0];
        float rj1 = Js[i][1] - Js[p][1];
        float rj2 = Js[i][2] - Js[p][2];
#pragma unroll
        for (int r = 0; r < 3; ++r) {
            float p0 = res[p][r * 4 + 0], p1 = res[p][r * 4 + 1], p2 = res[p][r * 4 + 2];
#pragma unroll
            for (int c2 = 0; c2 < 3; ++c2)
                res[i][r * 4 + c2] = p0 * R[0 * 3 + c2] + p1 * R[1 * 3 + c2] + p2 * R[2 * 3 + c2];
            res[i][r * 4 + 3] = p0 * rj0 + p1 * rj1 + p2 * rj2 + res[p][r * 4 + 3];
        }
    }

    float* A = ws + OFF_RELA + (size_t)b * (J_ * 16);
#pragma unroll
    for (int i = 0; i < J_; ++i) {
#pragma unroll
        for (int r = 0; r < 3; ++r) {
            float r0 = res[i][r * 4 + 0], r1 = res[i][r * 4 + 1], r2 = res[i][r * 4 + 2];
            A[i * 16 + r * 4 + 0] = r0;
            A[i * 16 + r * 4 + 1] = r1;
            A[i * 16 + r * 4 + 2] = r2;
            A[i * 16 + r * 4 + 3] = res[i][r * 4 + 3]
                - (r0 * Js[i][0] + r1 * Js[i][1] + r2 * Js[i][2]);
        }
        A[i * 16 + 12] = 0.f; A[i * 16 + 13] = 0.f; A[i * 16 + 14] = 0.f; A[i * 16 + 15] = 1.f;
    }
}

// =====================================================================
// 6) Fused WMMA GEMM: v_posed[b, n] = A[b, 0:220] . Dp[0:220, n] + tmpl[n]
//    - A tile (16x220) staged to LDS with global_load_async_to_lds_b128
//    - B fragment = one b64 load from the K-pair packed buffer
//    - 55 x v_wmma_f32_16x16x4_f32 per 16x16 output tile, branch-free
// =====================================================================
__global__ void k_gemm_wmma(const float* __restrict__ vtpl,
                            float* __restrict__ ws) {
    __shared__ float ldsA[16 * LDA];
    const int tid   = threadIdx.x;
    const int mBase = blockIdx.y * 16;

    // async-stage A tile [16 x 220] into LDS: 880 b128 transfers
    {
        const float* src = ws + OFF_ASTAGE + (size_t)mBase * KPAD;
        unsigned ldsBase = (unsigned)(uintptr_t)&ldsA[0];
        const int q4 = KPAD / 4;                       // 55 float4 per row
        for (int e = tid; e < 16 * q4; e += blockDim.x) {
            int r = e / q4, q = e - r * q4;
            unsigned dstOff = ldsBase + (unsigned)((r * LDA + q * 4) * 4);
            unsigned long long gaddr =
                (unsigned long long)(uintptr_t)(src + (size_t)r * KPAD + q * 4);
            asm volatile("global_load_async_to_lds_b128 %0, %1, off"
                         :: "v"(dstOff), "v"(gaddr) : "memory");
        }
        asm volatile("s_wait_asynccnt 0x0" ::: "memory");
    }
    __syncthreads();

    const int nTiles = (N3 + 15) / 16;                 // 1292
    const int nTile  = blockIdx.x * 4 + (tid >> 5);
    if (nTile >= nTiles) return;

    const int lane = tid & 31;
    const int half = lane >> 4;                        // K pair {0,1} vs {2,3}
    const int ml   = lane & 15;                        // A row / B column in tile
    const int n    = nTile * 16 + ml;
    const int nc   = (n < N3) ? n : (N3 - 1);          // clamp for tail tile

    const float* Bp = ws + OFF_DIRS + 2 * (size_t)nc;  // + pair*2*N3
    const float* Ap = &ldsA[ml * LDA];                 // + kloc

    v8f acc = {};
#pragma unroll 5
    for (int kk = 0; kk < KPAD; kk += 4) {
        const int kloc = kk + 2 * half;
        v2f a  = *(const v2f*)(Ap + kloc);                         // A[m][kloc..+1]
        v2f bf = *(const v2f*)(Bp + (size_t)(kloc >> 1) * (2 * N3)); // B[kloc..+1][n]
        acc = __builtin_amdgcn_wmma_f32_16x16x4_f32(
                  false, a, false, bf, (short)0, acc, false, false);
    }

    if (n < N3) {
        const float t = vtpl[n];                       // flat index v*3+c == n
        float* dst = ws + OFF_VPOSED;
#pragma unroll
        for (int r = 0; r < 8; ++r) {
            const int row = mBase + r + 8 * half;
            dst[(size_t)row * N3 + n] = acc[r] + t;
        }
    }
}

// =====================================================================
// 7) Skinning: rel_A[b] staged in LDS; T = lbs_w[v,:] @ rel_A[b]
// =====================================================================
__global__ void k_skin(const float* __restrict__ lbs,
                       const float* __restrict__ ws_ro,
                       float* __restrict__ out) {
    __shared__ float Asm[J_ * 16];
    const int b = blockIdx.y;
    for (int i = threadIdx.x; i < J_ * 16; i += blockDim.x)
        Asm[i] = ws_ro[OFF_RELA + (size_t)b * (J_ * 16) + i];
    __syncthreads();

    const int v = blockIdx.x * blockDim.x + threadIdx.x;
    if (v >= V_) return;

    float T[16];
#pragma unroll
    for (int i = 0; i < 16; ++i) T[i] = 0.f;
#pragma unroll
    for (int j = 0; j < J_; ++j) {
        const float w = lbs[(size_t)v * J_ + j];
#pragma unroll
        for (int i = 0; i < 16; ++i) T[i] += w * Asm[j * 16 + i];
    }

    const float* vp = ws_ro + OFF_VPOSED + (size_t)b * N3 + (size_t)v * 3;
    const float x = vp[0], y = vp[1], z = vp[2];
    float* o = out + OUT_VERT + ((size_t)b * V_ + v) * 3;
#pragma unroll
    for (int c = 0; c < 3; ++c)
        o[c] = T[c * 4 + 0] * x + T[c * 4 + 1] * y + T[c * 4 + 2] * z + T[c * 4 + 3];
}

// =====================================================================
// 8) Output joints: wave per (b,k)
// =====================================================================
__global__ void k_joints_out(const float* __restrict__ jreg,
                             float* __restrict__ out) {
    int unit = blockIdx.x * (blockDim.x >> 5) + (threadIdx.x >> 5);
    if (unit >= B_ * NK_) return;
    int b = unit / NK_, k = unit % NK_;
    int lane = threadIdx.x & 31;
    const float* vb = out + OUT_VERT + (size_t)b * V_ * 3;
    float a0 = 0.f, a1 = 0.f, a2 = 0.f;
    for (int v = lane; v < V_; v += 32) {
        float w = jreg[(size_t)v * NK_ + k];
        const float* p = vb + (size_t)v * 3;
        a0 += w * p[0]; a1 += w * p[1]; a2 += w * p[2];
    }
#pragma unroll
    for (int m = 16; m >= 1; m >>= 1) {
        a0 += __shfl_xor(a0, m, 32);
        a1 += __shfl_xor(a1, m, 32);
        a2 += __shfl_xor(a2, m, 32);
    }
    if (lane == 0) {
        float* d = out + OUT_JNT + (size_t)(b * NK_ + k) * 3;
        d[0] = a0; d[1] = a1; d[2] = a2;
    }
}

// =====================================================================
extern "C" void kernel_launch(void* const* d_in, const int* in_sizes, int n_in,
                              void* d_out, int out_size, void* d_ws, size_t ws_size,
                              hipStream_t stream) {
    const float* inputs    = (const float*)d_in[0];
    const float* vtpl      = (const float*)d_in[1];
    const float* shapedirs = (const float*)d_in[2];
    const float* sjreg     = (const float*)d_in[3];
    const float* posedirs  = (const float*)d_in[4];
    const float* lbs       = (const float*)d_in[5];
    const float* jreg      = (const float*)d_in[6];
    float* out = (float*)d_out;
    float* ws  = (float*)d_ws;

    // 1) A-matrix init (shape betas + zero pad)
    k_init_astage<<<(B_ * KPAD + 255) / 256, 256, 0, stream>>>(inputs, ws);
    // 2) rodrigues -> Rs + pose feature
    k_rodrigues<<<(B_ * J_ + 255) / 256, 256, 0, stream>>>(inputs, ws, out);
    // 3) pack Dirs into K-pair interleaved buffer
    {
        int total = (KPAD / 2) * N3;
        k_pack_dirs<<<(total + 255) / 256, 256, 0, stream>>>(shapedirs, posedirs, ws);
    }
    // 4) precontract joint regressor with shapedirs/template
    {
        int units = (NB_ + 1) * J_;
        k_joint_reduce<<<(units + 7) / 8, 256, 0, stream>>>(vtpl, shapedirs, sjreg, ws);
    }
    // 5) kinematic chain -> rel_A
    k_chain<<<(B_ + 63) / 64, 64, 0, stream>>>(inputs, out, ws);
    // 6) fused shape+pose WMMA GEMM -> v_posed
    {
        int nTiles = (N3 + 15) / 16;             // 1292
        dim3 grid((nTiles + 3) / 4, B_ / 16);    // (323, 64)
        k_gemm_wmma<<<grid, 128, 0, stream>>>(vtpl, ws);
    }
    // 7) skinning -> vertices
    {
        dim3 grid((V_ + 127) / 128, B_);
        k_skin<<<grid, 128, 0, stream>>>(lbs, ws, out);
    }
    // 8) output joints
    {
        int units = B_ * NK_;
        k_joints_out<<<(units + 7) / 8, 256, 0, stream>>>(jreg, out);
    }
}